// GaussianBluerConv2_68891275428282
// MI455X (gfx1250) — compile-verified
//
#include <hip/hip_runtime.h>

typedef float v2f __attribute__((ext_vector_type(2)));
typedef float v8f __attribute__((ext_vector_type(8)));

#define Bn 16
#define Cn 256
#define Hn 128
#define Wn 128
#define HW (Hn * Wn)          // 16384
#define CHW (Cn * HW)         // 4194304

// ---------------------------------------------------------------------------
// Kernel 1: s[b,h,w] = sum_c x[b,c,h,w]
// Mapped onto V_WMMA_F32_16X16X4_F32 with A = all-ones:
//   D[m,n] += sum_k A[m,k]*B[k,n] = sum_k B[k,n]
// Each wave owns 16 consecutive pixels (the N dimension) and accumulates the
// 256-channel sum in 64 WMMA steps of K=4 channels. Since A is all-ones, any
// assignment of the 4 channels to B's (VGPR, lane-half) slots yields the same
// sum, so the operand striping is correctness-neutral for this reduction.
// ---------------------------------------------------------------------------
__global__ __launch_bounds__(256) void sum_channels_wmma(
    const float* __restrict__ x, float* __restrict__ s) {
    const int lane = threadIdx.x & 31;           // wave32
    const int wave = threadIdx.x >> 5;           // 8 waves / block
    const int chunk = blockIdx.x * 8 + wave;     // 16 pixels per wave
    const int P = chunk * 16;                    // flat pixel base = b*HW + q
    const int b = P / HW;
    const int q = P - b * HW;                    // HW % 16 == 0 -> no straddle
    const int n  = lane & 15;                    // pixel within chunk (N)
    const int kh = lane >> 4;                    // lane half selects K pair

    const float* base = x + (size_t)b * CHW + (size_t)(q + n);

    v2f ones; ones[0] = 1.0f; ones[1] = 1.0f;    // A matrix: all ones
    v8f acc = {};

    #pragma unroll 4
    for (int c = 0; c < Cn; c += 4) {
        v2f bm;
        bm[0] = base[(size_t)(c + 2 * kh)     * HW];
        bm[1] = base[(size_t)(c + 2 * kh + 1) * HW];
        // (neg_a, A, neg_b, B, c_mod, C, reuse_a, reuse_b)
        acc = __builtin_amdgcn_wmma_f32_16x16x4_f32(
            false, ones, false, bm, (short)0, acc, false, false);
    }

    // D layout: VGPR0, lanes 0-15 = row M=0, N=lane (all rows identical here)
    if (lane < 16) s[P + lane] = acc[0];
}

// ---------------------------------------------------------------------------
// Kernel 2: x1 = conv3x3(s), zero padding 1 (cross-correlation, as lax.conv).
// s and x1 are 1 MiB each -> cache-resident, negligible cost.
// ---------------------------------------------------------------------------
__global__ __launch_bounds__(256) void conv3x3(
    const float* __restrict__ s, const float* __restrict__ f,
    float* __restrict__ x1) {
    const int i = blockIdx.x * blockDim.x + threadIdx.x;   // [0, Bn*HW)
    const int b = i >> 14;            // / 16384
    const int q = i & (HW - 1);
    const int h = q >> 7;
    const int w = q & (Wn - 1);

    const float* sb = s + (size_t)b * HW;
    float acc = 0.0f;
    #pragma unroll
    for (int dy = 0; dy < 3; ++dy) {
        const int hh = h + dy - 1;
        if (hh < 0 || hh >= Hn) continue;
        #pragma unroll
        for (int dx = 0; dx < 3; ++dx) {
            const int ww = w + dx - 1;
            if (ww < 0 || ww >= Wn) continue;
            acc += f[dy * 3 + dx] * sb[hh * Wn + ww];
        }
    }
    x1[i] = acc;
}

// ---------------------------------------------------------------------------
// Kernel 3: out[b,c,h,w] = x[b,c,h,w] - x1[b,h,w]  (float4 streaming pass)
// This is the dominant 512 MB of traffic; x1 re-reads hit L2 (1 MiB resident).
// ---------------------------------------------------------------------------
__global__ __launch_bounds__(256) void subtract_bcast(
    const float* __restrict__ x, const float* __restrict__ x1,
    float* __restrict__ out) {
    const size_t i4 = (size_t)blockIdx.x * blockDim.x + threadIdx.x;
    const size_t i  = i4 * 4;                       // element index
    const size_t b  = i >> 22;                      // / (Cn*HW)
    const size_t q  = i & (size_t)(HW - 1);         // pixel offset in image

    const float4 xv = *(const float4*)(x  + i);
    const float4 sv = *(const float4*)(x1 + b * HW + q);
    float4 ov;
    ov.x = xv.x - sv.x;
    ov.y = xv.y - sv.y;
    ov.z = xv.z - sv.z;
    ov.w = xv.w - sv.w;
    *(float4*)(out + i) = ov;
}

extern "C" void kernel_launch(void* const* d_in, const int* in_sizes, int n_in,
                              void* d_out, int out_size, void* d_ws, size_t ws_size,
                              hipStream_t stream) {
    const float* x = (const float*)d_in[0];       // (16,256,128,128) f32
    const float* f = (const float*)d_in[1];       // (3,3) f32
    float* out = (float*)d_out;

    float* s  = (float*)d_ws;                                  // Bn*HW f32 = 1 MiB
    float* x1 = (float*)((char*)d_ws + (size_t)Bn * HW * 4);   // Bn*HW f32 = 1 MiB

    // Kernel 1: 16384 wave-chunks of 16 pixels, 8 waves per 256-thread block
    sum_channels_wmma<<<(Bn * HW / 16) / 8, 256, 0, stream>>>(x, s);

    // Kernel 2: one thread per (b,h,w)
    conv3x3<<<(Bn * HW) / 256, 256, 0, stream>>>(s, f, x1);

    // Kernel 3: float4 per thread over all Bn*Cn*HW elements
    subtract_bcast<<<(Bn * (size_t)Cn * HW / 4) / 256, 256, 0, stream>>>(x, x1, out);
}